// ProbAttention_53403623358558
// MI455X (gfx1250) — compile-verified
//
#include <hip/hip_runtime.h>
#include <math.h>

#define B_   4
#define L_   2048
#define H_   8
#define D_   64
#define U_   40     // top-k queries (FACTOR*ceil(ln(2048)) = 40)
#define SAMP 40     // sampled keys per query
#define UPAD 48     // 3 WMMA M-tiles
#define NT_  (L_ / 16)

typedef __attribute__((ext_vector_type(2))) float v2f;
typedef __attribute__((ext_vector_type(8))) float v8f;

// D = A(16x4,f32) * B(4x16,f32) + C(16x16,f32)  -- full f32 matrix pipe
#define WMMA_F32(a, b, c) \
  __builtin_amdgcn_wmma_f32_16x16x4_f32(false, (a), false, (b), (short)0, (c), false, false)

// ---------------------------------------------------------------------------
// JAX threefry2x32, key = jax.random.key(42) -> (0, 42).
// random_bits for flat size 81920: counts split in halves (x0=i, x1=i+40960),
// bits[i] = out0 if i < 40960 else out1.  randint span 2048: multiplier
// (2^16 % 2048)^2 % 2048 == 0  ->  idx = lower16 % 2048 = bits & 0x7FF.
// ---------------------------------------------------------------------------
__device__ __forceinline__ unsigned rotl32(unsigned x, int r) {
  return (x << r) | (x >> (32 - r));
}

__device__ __forceinline__ unsigned threefry_sample(unsigned i) {
  const unsigned half = (L_ * SAMP) / 2;  // 40960
  const unsigned p = (i < half) ? i : (i - half);
  unsigned x0 = p, x1 = p + half;
  const unsigned ks[3] = {0u, 42u, 0u ^ 42u ^ 0x1BD11BDAu};
  x0 += ks[0];
  x1 += ks[1];
  const int rots[8] = {13, 15, 26, 6, 17, 29, 16, 24};
#pragma unroll
  for (int g = 0; g < 5; ++g) {
#pragma unroll
    for (int rr = 0; rr < 4; ++rr) {
      const int r = rots[(g & 1) * 4 + rr];
      x0 += x1;
      x1 = rotl32(x1, r);
      x1 ^= x0;
    }
    x0 += ks[(g + 1) % 3];
    x1 += ks[(g + 2) % 3] + (unsigned)(g + 1);
  }
  const unsigned bits = (i < half) ? x0 : x1;
  return bits & 2047u;
}

// ---------------------------------------------------------------------------
// Kernel 1: sparsity measure M[b,h,l] = max_s(q.k_s) - sum_s(q.k_s)/2048
// ---------------------------------------------------------------------------
__global__ __launch_bounds__(256) void probe_scores_kernel(
    const float* __restrict__ Q, const float* __restrict__ K,
    float* __restrict__ Mout) {
  const int t = blockIdx.x * 256 + threadIdx.x;
  if (t >= B_ * H_ * L_) return;
  const int l = t & (L_ - 1);
  const int h = (t >> 11) & (H_ - 1);
  const int b = t >> 14;

  const float4* qp = (const float4*)(Q + (((size_t)b * L_ + l) * H_ + h) * D_);
  float4 qr[16];
#pragma unroll
  for (int c = 0; c < 16; ++c) qr[c] = qp[c];

  float mx = -__builtin_inff();
  float sum = 0.0f;
  for (int s = 0; s < SAMP; ++s) {
    const unsigned idx = threefry_sample((unsigned)(l * SAMP + s));
    const float4* kp =
        (const float4*)(K + (((size_t)b * L_ + idx) * H_ + h) * D_);
    float acc = 0.0f;
#pragma unroll
    for (int c = 0; c < 16; ++c) {
      const float4 kv = kp[c];
      acc += qr[c].x * kv.x + qr[c].y * kv.y + qr[c].z * kv.z + qr[c].w * kv.w;
    }
    mx = fmaxf(mx, acc);
    sum += acc;
  }
  Mout[((b * H_ + h) << 11) + l] = mx - sum * (1.0f / (float)L_);
}

// ---------------------------------------------------------------------------
// Kernel 2: top-40 indices per (b,h); ties -> lowest index (lax.top_k).
// ---------------------------------------------------------------------------
__global__ __launch_bounds__(256) void topk_kernel(const float* __restrict__ M,
                                                   int* __restrict__ top) {
  const int bh = blockIdx.x;
  __shared__ float vals[L_];
  __shared__ float rv[256];
  __shared__ int ri[256];

  for (int i = threadIdx.x; i < L_; i += 256) vals[i] = M[bh * L_ + i];
  __syncthreads();

  for (int it = 0; it < U_; ++it) {
    float best = -__builtin_inff();
    int bi = L_;
    for (int i = threadIdx.x; i < L_; i += 256) {
      const float v = vals[i];
      if (v > best || (v == best && i < bi)) { best = v; bi = i; }
    }
    rv[threadIdx.x] = best;
    ri[threadIdx.x] = bi;
    __syncthreads();
    for (int s = 128; s > 0; s >>= 1) {
      if (threadIdx.x < (unsigned)s) {
        const float ov = rv[threadIdx.x + s];
        const int oi = ri[threadIdx.x + s];
        if (ov > rv[threadIdx.x] ||
            (ov == rv[threadIdx.x] && oi < ri[threadIdx.x])) {
          rv[threadIdx.x] = ov;
          ri[threadIdx.x] = oi;
        }
      }
      __syncthreads();
    }
    if (threadIdx.x == 0) {
      top[bh * U_ + it] = ri[0];
      vals[ri[0]] = -__builtin_inff();
    }
    __syncthreads();
  }
}

// ---------------------------------------------------------------------------
// Kernel 3: V column means per (b,h)
// ---------------------------------------------------------------------------
__global__ __launch_bounds__(256) void vmean_kernel(const float* __restrict__ V,
                                                    float* __restrict__ meanOut) {
  const int bh = blockIdx.x;
  const int b = bh >> 3, h = bh & 7;
  const int d = threadIdx.x & 63;
  const int part = threadIdx.x >> 6;  // 0..3
  float acc = 0.0f;
  for (int l = part; l < L_; l += 4)
    acc += V[(((size_t)b * L_ + l) * H_ + h) * D_ + d];
  __shared__ float red[256];
  red[threadIdx.x] = acc;
  __syncthreads();
  if (part == 0) {
    const float v = red[d] + red[d + 64] + red[d + 128] + red[d + 192];
    meanOut[bh * D_ + d] = v * (1.0f / (float)L_);
  }
}

// ---------------------------------------------------------------------------
// Kernel 4: broadcast V_mean into the whole (B,L,H,D) output
// ---------------------------------------------------------------------------
__global__ __launch_bounds__(256) void fill_kernel(const float* __restrict__ mean,
                                                   float* __restrict__ out) {
  const int i = blockIdx.x * 256 + threadIdx.x;
  const int n4 = (B_ * L_ * H_ * D_) / 4;
  if (i >= n4) return;
  const int e = i * 4;
  const int b = e >> 20;          // L_*H_*D_ = 2^20 per batch
  const int h = (e >> 6) & 7;
  const int d = e & 63;
  *(float4*)(out + e) = *(const float4*)(mean + (b * H_ + h) * D_ + d);
}

// ---------------------------------------------------------------------------
// Kernel 5: dense attention for the top-40 queries, flash-style online
// softmax, both GEMMs on V_WMMA_F32_16X16X4_F32.  K/V tiles stream into a
// double-buffered LDS via CDNA5 async global->LDS loads (ASYNCcnt), so the
// next tile's fill overlaps the current tile's WMMAs.
// Block = 128 threads (4 waves) per (b,h); waves 0..2 own M-tiles 0..2.
// ---------------------------------------------------------------------------
__global__ __launch_bounds__(128) void attn_scatter_kernel(
    const float* __restrict__ Q, const float* __restrict__ K,
    const float* __restrict__ V, const int* __restrict__ top,
    float* __restrict__ out) {
  __shared__ float Klds[2][16 * 65];    // padded stride 65 -> no bank conflicts
  __shared__ float Vlds[2][16 * 65];
  __shared__ float Plds[3][16 * 17];    // per-wave P tile, C-layout -> A-layout
  __shared__ int qidx[UPAD];

  const int bh = blockIdx.x;
  const int b = bh >> 3, h = bh & 7;
  const int wave = threadIdx.x >> 5;
  const int lane = threadIdx.x & 31;
  const int lmod = lane & 15;
  const int lhi = lane >> 4;

  if (threadIdx.x < UPAD)
    qidx[threadIdx.x] = (threadIdx.x < U_) ? top[bh * U_ + threadIdx.x] : -1;
  __syncthreads();

  // Per-thread tile-slice coordinates: 128 threads x 2 x float4 cover a
  // 16x64 tile.  Issue async global->LDS B128 loads (no VGPR round-trip).
  auto issue_tile = [&](int kt, int buf) {
#pragma unroll
    for (int it = 0; it < 2; ++it) {
      const int v4 = (int)threadIdx.x + it * 128;  // 0..255
      const int r = v4 >> 4;
      const int c = (v4 & 15) * 4;
      const size_t gofs = (((size_t)b * L_ + kt * 16 + r) * H_ + h) * D_ + c;
      const unsigned kl = (unsigned)(size_t)&Klds[buf][r * 65 + c];
      const unsigned vl = (unsigned)(size_t)&Vlds[buf][r * 65 + c];
      const float* kg = K + gofs;
      const float* vg = V + gofs;
      asm volatile("global_load_async_to_lds_b128 %0, %1, off"
                   :: "v"(kl), "v"(kg) : "memory");
      asm volatile("global_load_async_to_lds_b128 %0, %1, off"
                   :: "v"(vl), "v"(vg) : "memory");
    }
  };

  // A fragments for Q tile: lane holds row m = lane&15; frag s covers
  // K-dims {4s+2*lhi, 4s+2*lhi+1}. Padded rows (q<0) are zero.
  v2f aq[16];
  if (wave < 3) {
    const int q = qidx[wave * 16 + lmod];
    if (q >= 0) {
      const float* qp = Q + (((size_t)b * L_ + q) * H_ + h) * D_;
#pragma unroll
      for (int s = 0; s < 16; ++s) {
        const int k0 = s * 4 + 2 * lhi;
        aq[s].x = qp[k0];
        aq[s].y = qp[k0 + 1];
      }
    } else {
#pragma unroll
      for (int s = 0; s < 16; ++s) { aq[s].x = 0.0f; aq[s].y = 0.0f; }
    }
  }

  v8f o[4];
  v8f mrow, lrow;
#pragma unroll
  for (int t = 0; t < 4; ++t) o[t] = (v8f){};
#pragma unroll
  for (int i = 0; i < 8; ++i) { mrow[i] = -__builtin_inff(); lrow[i] = 0.0f; }

  // Prologue: stream tile 0 into buffer 0.
  issue_tile(0, 0);
  asm volatile("s_wait_asynccnt 0x0" ::: "memory");
  __syncthreads();

  for (int kt = 0; kt < NT_; ++kt) {
    const int cur = kt & 1;
    if (kt + 1 < NT_) issue_tile(kt + 1, cur ^ 1);  // overlap fill w/ compute

    if (wave < 3) {
      const float* kl = &Klds[cur][0];
      const float* vl = &Vlds[cur][0];
      // S = Q_tile * K_tile^T : B[k][n] = K[n][k]
      v8f s = (v8f){};
#pragma unroll
      for (int st = 0; st < 16; ++st) {
        const int k0 = st * 4 + 2 * lhi;
        v2f bk;
        bk.x = kl[lmod * 65 + k0];
        bk.y = kl[lmod * 65 + k0 + 1];
        s = WMMA_F32(aq[st], bk, s);
      }
      // scale + online softmax in C-layout (row = v + 8*lhi over 16 lanes)
      v8f tmax;
#pragma unroll
      for (int i = 0; i < 8; ++i) { s[i] *= 0.125f; tmax[i] = s[i]; }
#pragma unroll
      for (int off = 1; off < 16; off <<= 1)
#pragma unroll
        for (int i = 0; i < 8; ++i)
          tmax[i] = fmaxf(tmax[i], __shfl_xor(tmax[i], off, 32));

      v8f mnew, alpha, p, psum;
#pragma unroll
      for (int i = 0; i < 8; ++i) {
        mnew[i] = fmaxf(mrow[i], tmax[i]);
        alpha[i] = __expf(mrow[i] - mnew[i]);  // first tile: exp(-inf)=0
        p[i] = __expf(s[i] - mnew[i]);
        psum[i] = p[i];
      }
#pragma unroll
      for (int off = 1; off < 16; off <<= 1)
#pragma unroll
        for (int i = 0; i < 8; ++i) psum[i] += __shfl_xor(psum[i], off, 32);
#pragma unroll
      for (int i = 0; i < 8; ++i) {
        lrow[i] = lrow[i] * alpha[i] + psum[i];
        mrow[i] = mnew[i];
      }
#pragma unroll
      for (int t = 0; t < 4; ++t)
#pragma unroll
        for (int i = 0; i < 8; ++i) o[t][i] *= alpha[i];

      // Stage P through LDS to re-lay C-layout as an A-matrix.
      float* pl = &Plds[wave][0];
#pragma unroll
      for (int i = 0; i < 8; ++i) pl[(i + 8 * lhi) * 17 + lmod] = p[i];
      asm volatile("s_wait_dscnt 0x0" ::: "memory");  // in-wave LDS RAW

      // O += P * V_tile  (K-dim 16 -> 4 WMMA steps; 4 N-subtiles of 16 cols)
#pragma unroll
      for (int kc = 0; kc < 4; ++kc) {
        const int kk = kc * 4 + 2 * lhi;
        v2f ap;
        ap.x = pl[lmod * 17 + kk];
        ap.y = pl[lmod * 17 + kk + 1];
#pragma unroll
        for (int t = 0; t < 4; ++t) {
          const int d = t * 16 + lmod;
          v2f bv;
          bv.x = vl[kk * 65 + d];
          bv.y = vl[(kk + 1) * 65 + d];
          o[t] = WMMA_F32(ap, bv, o[t]);
        }
      }
    }

    // Next buffer fully resident + current buffer fully consumed.
    asm volatile("s_wait_asynccnt 0x0" ::: "memory");
    __syncthreads();
  }

  // Normalize and scatter valid rows: out[b, q, h, d] = O[m][d] / l[m]
  if (wave < 3) {
#pragma unroll
    for (int i = 0; i < 8; ++i) {
      const int row = wave * 16 + i + 8 * lhi;
      if (row < U_) {
        const int q = qidx[row];
        const float inv = 1.0f / lrow[i];
#pragma unroll
        for (int t = 0; t < 4; ++t) {
          const int d = t * 16 + lmod;
          out[(((size_t)b * L_ + q) * H_ + h) * D_ + d] = o[t][i] * inv;
        }
      }
    }
  }
}

// ---------------------------------------------------------------------------
extern "C" void kernel_launch(void* const* d_in, const int* in_sizes, int n_in,
                              void* d_out, int out_size, void* d_ws,
                              size_t ws_size, hipStream_t stream) {
  (void)in_sizes; (void)n_in; (void)out_size; (void)ws_size;
  const float* Q = (const float*)d_in[0];
  const float* K = (const float*)d_in[1];
  const float* V = (const float*)d_in[2];
  float* out = (float*)d_out;

  // workspace layout: M (64K f32) | top (1280 i32, padded 8KB) | mean (2K f32)
  float* wsM = (float*)d_ws;
  int* wsTop = (int*)((char*)d_ws + (size_t)(B_ * H_ * L_) * 4);
  float* wsMean = (float*)((char*)d_ws + (size_t)(B_ * H_ * L_) * 4 + 8192);

  probe_scores_kernel<<<(B_ * H_ * L_ + 255) / 256, 256, 0, stream>>>(Q, K, wsM);
  topk_kernel<<<B_ * H_, 256, 0, stream>>>(wsM, wsTop);
  vmean_kernel<<<B_ * H_, 256, 0, stream>>>(V, wsMean);
  fill_kernel<<<(B_ * L_ * H_ * D_ / 4 + 255) / 256, 256, 0, stream>>>(wsMean, out);
  attn_scatter_kernel<<<B_ * H_, 128, 0, stream>>>(Q, K, V, wsTop, out);
}